// GraphResNet_88510686036733
// MI455X (gfx1250) — compile-verified
//
#include <hip/hip_runtime.h>
#include <hip/hip_bf16.h>

typedef __attribute__((ext_vector_type(2))) float v2f;
typedef __attribute__((ext_vector_type(8))) float v8f;

#define CH        128
#define HEADS     4
#define CPH       32
#define NEG_SLOPE 0.2f
#define BN_EPS    1e-5f

// ---------------------------------------------------------------------------
// GEMM: H[N,128] = X[N,128] @ W[128,128] via V_WMMA_F32_16X16X4_F32.
// Block = 256 threads = 8 waves; block b owns M-tile b (16 rows); wave w owns
// N-tile w (16 cols). The 16x128 X tile (8 KB) is staged once in LDS and
// shared by all 8 waves (A fragments come from ds_load); B fragments come
// straight from global (W is L2-resident; per-block traffic identical either
// way). K-loop: 128/4 = 32 chained WMMA accumulations, unrolled 8x.
// A frag (16x4 f32, 2 VGPR): lane l -> M = l&15, K = (l>>4)*2 + {0,1}
// B frag (4x16 f32, 2 VGPR): lane l -> N = l&15, K = (l>>4)*2 + {0,1}
// C/D (16x16 f32, 8 VGPR):   lane l -> N = l&15, M = r + (l>>4)*8
// ---------------------------------------------------------------------------
__global__ void k_gemm_wmma(const float* __restrict__ X,
                            const float* __restrict__ W,
                            float* __restrict__ H, int n_nodes) {
  __shared__ float xs[16 * CH];   // 8 KB X tile

  const int mtile = blockIdx.x;
  const int wave  = threadIdx.x >> 5;
  const int lane  = threadIdx.x & 31;
  const int m0 = mtile << 4;
  const int n0 = wave  << 4;
  const int lr = lane & 15;
  const int kb = (lane >> 4) << 1;

  // cooperative stage of the 16x128 A tile (2048 floats / 256 threads = 8 ea)
  #pragma unroll
  for (int i = threadIdx.x; i < 16 * CH; i += 256) {
    int row = m0 + (i >> 7);
    if (row >= n_nodes) row = n_nodes - 1;     // clamp (N%16==0 here anyway)
    xs[i] = X[(size_t)row * CH + (i & (CH - 1))];
  }
  __syncthreads();

  v8f acc = {0.f, 0.f, 0.f, 0.f, 0.f, 0.f, 0.f, 0.f};
  const float* __restrict__ arow = xs + lr * CH;

  #pragma unroll 8
  for (int k = 0; k < CH; k += 4) {
    v2f a, b;
    a.x = arow[k + kb + 0];
    a.y = arow[k + kb + 1];
    b.x = W[(size_t)(k + kb + 0) * CH + n0 + lr];
    b.y = W[(size_t)(k + kb + 1) * CH + n0 + lr];
    acc = __builtin_amdgcn_wmma_f32_16x16x4_f32(
        /*neg_a=*/false, a, /*neg_b=*/false, b,
        /*c_mod=*/(short)0, acc, /*reuse_a=*/false, /*reuse_b=*/false);
  }

  const int nn = n0 + lr;
  const int mb = (lane >> 4) << 3;
  if (m0 + 16 <= n_nodes) {                    // uniform fast path: no guards
    #pragma unroll
    for (int r = 0; r < 8; ++r)
      H[(size_t)(m0 + mb + r) * CH + nn] = acc[r];
  } else {
    #pragma unroll
    for (int r = 0; r < 8; ++r) {
      const int m = m0 + mb + r;
      if (m < n_nodes) H[(size_t)m * CH + nn] = acc[r];
    }
  }
}

// ---------------------------------------------------------------------------
// Per-node attention logits: alpha_src/dst[n,h] = sum_c H[n,h*32+c]*att[h,c]
// ---------------------------------------------------------------------------
__global__ void k_alpha(const float* __restrict__ H,
                        const float* __restrict__ aw_s,
                        const float* __restrict__ aw_d,
                        float* __restrict__ as_n, float* __restrict__ ad_n,
                        int n_nodes) {
  const int t = blockIdx.x * blockDim.x + threadIdx.x;
  if (t >= n_nodes * HEADS) return;
  const int n = t >> 2, h = t & 3;
  const float* __restrict__ hp = H + (size_t)n * CH + h * CPH;
  const float* __restrict__ sw = aw_s + h * CPH;
  const float* __restrict__ dw = aw_d + h * CPH;
  float s = 0.f, d = 0.f;
  #pragma unroll
  for (int c = 0; c < CPH; ++c) { const float v = hp[c]; s += v * sw[c]; d += v * dw[c]; }
  as_n[t] = s;
  ad_n[t] = d;
}

// ---------------------------------------------------------------------------
// Zero/neg-inf initialization of per-layer scratch.
// ---------------------------------------------------------------------------
__global__ void k_init(float* __restrict__ agg, float* __restrict__ emax,
                       float* __restrict__ denom, float* __restrict__ stats,
                       long total_feat, int n4) {
  const long stride = (long)gridDim.x * blockDim.x;
  for (long i = (long)blockIdx.x * blockDim.x + threadIdx.x; i < total_feat; i += stride) {
    agg[i] = 0.f;
    if (i < n4)  { emax[i] = -3.0e38f; denom[i] = 0.f; }
    if (i < 512) stats[i] = 0.f;
  }
}

__device__ inline void atomicMaxF(float* addr, float val) {
  int* ia = reinterpret_cast<int*>(addr);
  int old = __float_as_int(*(volatile float*)addr);
  while (__int_as_float(old) < val) {
    const int prev = atomicCAS(ia, old, __float_as_int(val));
    if (prev == old) break;
    old = prev;
  }
}

__device__ inline float lrelu(float v) { return v > 0.f ? v : NEG_SLOPE * v; }

// segment max of leaky_relu(alpha_src[src]+alpha_dst[dst]) over dst
__global__ void k_edge_max(const int* __restrict__ ei,
                           const float* __restrict__ as, const float* __restrict__ ad,
                           float* __restrict__ emax, int E, int N) {
  const int t = blockIdx.x * blockDim.x + threadIdx.x;
  if (t >= (E + N) * HEADS) return;
  const int e = t >> 2, h = t & 3;
  const int s = (e < E) ? ei[e]     : (e - E);
  const int d = (e < E) ? ei[E + e] : (e - E);
  atomicMaxF(&emax[d * HEADS + h], lrelu(as[s * HEADS + h] + ad[d * HEADS + h]));
}

// ee = exp(e - emax[dst]); stash per-edge ee; segment-sum into denom
__global__ void k_edge_sum(const int* __restrict__ ei,
                           const float* __restrict__ as, const float* __restrict__ ad,
                           const float* __restrict__ emax,
                           float* __restrict__ ee_out, float* __restrict__ denom,
                           int E, int N) {
  const int t = blockIdx.x * blockDim.x + threadIdx.x;
  if (t >= (E + N) * HEADS) return;
  const int e = t >> 2, h = t & 3;
  const int s = (e < E) ? ei[e]     : (e - E);
  const int d = (e < E) ? ei[E + e] : (e - E);
  const float v  = lrelu(as[s * HEADS + h] + ad[d * HEADS + h]);
  const float ee = expf(v - emax[d * HEADS + h]);
  ee_out[t] = ee;
  atomicAdd(&denom[d * HEADS + h], ee);
}

// agg[dst,c] += h[src,c] * (ee / denom[dst,head(c)])  -- the hot scatter
__global__ void k_scatter(const int* __restrict__ ei, const float* __restrict__ H,
                          const float* __restrict__ ee, const float* __restrict__ denom,
                          float* __restrict__ agg, int E, int N) {
  const long total  = (long)(E + N) * CH;
  const long stride = (long)gridDim.x * blockDim.x;
  for (long i = (long)blockIdx.x * blockDim.x + threadIdx.x; i < total; i += stride) {
    const int e = (int)(i >> 7);
    const int c = (int)(i & (CH - 1));
    const int h = c >> 5;
    const int s = (e < E) ? ei[e]     : (e - E);
    const int d = (e < E) ? ei[E + e] : (e - E);
    const float coef = ee[e * HEADS + h] / denom[d * HEADS + h];
    atomicAdd(&agg[(size_t)d * CH + c], H[(size_t)s * CH + c] * coef);
  }
}

// ---------------------------------------------------------------------------
// BN statistics. NOTE: conv bias is absorbed exactly by BN (constant per-
// channel shift cancels in (y - mean) and leaves var unchanged), so we run
// BN directly on the aggregated output and never touch d_in[5].
// ---------------------------------------------------------------------------
__global__ void k_bn_stats(const float* __restrict__ Y, float* __restrict__ stats,
                           int n_nodes) {
  __shared__ float ss[256];
  __shared__ float sq[256];
  const int c    = threadIdx.x & (CH - 1);
  const int half = threadIdx.x >> 7;          // 0/1: two threads per channel
  const int r0 = blockIdx.x * 128;
  int r1 = r0 + 128; if (r1 > n_nodes) r1 = n_nodes;
  float s = 0.f, q = 0.f;
  for (int r = r0 + half; r < r1; r += 2) {
    const float v = Y[(size_t)r * CH + c];
    s += v; q += v * v;
  }
  ss[threadIdx.x] = s; sq[threadIdx.x] = q;
  __syncthreads();
  if (half == 0) {
    atomicAdd(&stats[c],      ss[threadIdx.x] + ss[threadIdx.x + 128]);
    atomicAdd(&stats[CH + c], sq[threadIdx.x] + sq[threadIdx.x + 128]);
  }
}

__global__ void k_bn_finalize(const float* __restrict__ stats,
                              const float* __restrict__ gamma,
                              const float* __restrict__ beta,
                              float* __restrict__ scale, float* __restrict__ shift,
                              float inv_n) {
  const int c = threadIdx.x;
  const float mu  = stats[c] * inv_n;
  const float var = stats[CH + c] * inv_n - mu * mu;
  const float sc  = gamma[c] * rsqrtf(var + BN_EPS);
  scale[c] = sc;
  shift[c] = beta[c] - mu * sc;
}

// out = relu( bn(agg) [+ resid] )   (every layer in the net ends in ReLU)
__global__ void k_bn_apply(const float* __restrict__ Y,
                           const float* __restrict__ scale,
                           const float* __restrict__ shift,
                           const float* __restrict__ resid,
                           float* __restrict__ out, long total) {
  const long stride = (long)gridDim.x * blockDim.x;
  for (long i = (long)blockIdx.x * blockDim.x + threadIdx.x; i < total; i += stride) {
    const int c = (int)(i & (CH - 1));
    float v = Y[i] * scale[c] + shift[c];
    if (resid) v += resid[i];
    out[i] = fmaxf(v, 0.f);
  }
}

// ---------------------------------------------------------------------------
extern "C" void kernel_launch(void* const* d_in, const int* in_sizes, int n_in,
                              void* d_out, int out_size, void* d_ws, size_t ws_size,
                              hipStream_t stream) {
  (void)n_in; (void)out_size; (void)ws_size;
  const float* x       = (const float*)d_in[0];
  const int*   ei      = (const int*)  d_in[1];
  const float* Ws      = (const float*)d_in[2];
  const float* att_src = (const float*)d_in[3];
  const float* att_dst = (const float*)d_in[4];
  // d_in[5] (biases) intentionally unused: exactly absorbed by BatchNorm.
  const float* gammas  = (const float*)d_in[6];
  const float* betas   = (const float*)d_in[7];
  float* out = (float*)d_out;

  const int N = in_sizes[0] / CH;       // 50000
  const int E = in_sizes[1] / 2;        // 600000
  const int ETOT = E + N;               // + self loops
  const size_t feat = (size_t)N * CH;

  // workspace carve-out (256B aligned)
  char* wp = (char*)d_ws;
  auto carve = [&](size_t bytes) -> float* {
    float* p = (float*)wp;
    wp += (bytes + 255) & ~(size_t)255;
    return p;
  };
  float* H     = carve(feat * sizeof(float));
  float* AGG   = carve(feat * sizeof(float));
  float* F0    = carve(feat * sizeof(float));
  float* F1    = carve(feat * sizeof(float));
  float* AS    = carve((size_t)N * HEADS * sizeof(float));
  float* AD    = carve((size_t)N * HEADS * sizeof(float));
  float* EMAX  = carve((size_t)N * HEADS * sizeof(float));
  float* DENOM = carve((size_t)N * HEADS * sizeof(float));
  float* COEF  = carve((size_t)ETOT * HEADS * sizeof(float));
  float* STATS = carve(512 * sizeof(float)); // [0,128)=sum [128,256)=sumsq [256,384)=scale [384,512)=shift

  // layer schedule: stem L0,L1; block0 L2,L3(+res F1); block1 L4,L5(+res F0)
  const float* ins[6]    = { x,  F0, F1, F0, F0, F1 };
  float*       outs[6]   = { F0, F1, F0, F0, F1, out };
  const float* resids[6] = { nullptr, nullptr, nullptr, F1, nullptr, F0 };

  const int gemm_blocks  = (N + 15) / 16;
  const int alpha_blocks = (N * HEADS + 255) / 256;
  const int edge_blocks  = (ETOT * HEADS + 255) / 256;
  const int stat_blocks  = (N + 127) / 128;

  for (int i = 0; i < 6; ++i) {
    const float* Wl = Ws      + (size_t)i * CH * CH;
    const float* Sl = att_src + (size_t)i * HEADS * CPH;
    const float* Dl = att_dst + (size_t)i * HEADS * CPH;

    k_gemm_wmma<<<gemm_blocks, 256, 0, stream>>>(ins[i], Wl, H, N);
    k_init<<<2048, 256, 0, stream>>>(AGG, EMAX, DENOM, STATS, (long)feat, N * HEADS);
    k_alpha<<<alpha_blocks, 256, 0, stream>>>(H, Sl, Dl, AS, AD, N);
    k_edge_max<<<edge_blocks, 256, 0, stream>>>(ei, AS, AD, EMAX, E, N);
    k_edge_sum<<<edge_blocks, 256, 0, stream>>>(ei, AS, AD, EMAX, COEF, DENOM, E, N);
    k_scatter<<<4096, 256, 0, stream>>>(ei, H, COEF, DENOM, AGG, E, N);
    k_bn_stats<<<stat_blocks, 256, 0, stream>>>(AGG, STATS, N);
    k_bn_finalize<<<1, CH, 0, stream>>>(STATS, gammas + (size_t)i * CH, betas + (size_t)i * CH,
                                        STATS + 256, STATS + 384, 1.0f / (float)N);
    k_bn_apply<<<2048, 256, 0, stream>>>(AGG, STATS + 256, STATS + 384, resids[i],
                                         outs[i], (long)feat);
  }
}